// MambaTemp_14405320311406
// MI455X (gfx1250) — compile-verified
//
#include <hip/hip_runtime.h>

// ---------------------------------------------------------------------------
// Mamba block for MI455X (gfx1250, wave32).
// Shapes: B=8, T=8, L=196, D=E=192, C=E*T=1536, N=16, Kconv=4, R=96.
// ---------------------------------------------------------------------------

#define BSZ   8
#define T_    8
#define L_    196
#define D_    192
#define E_    192
#define C_    1536
#define N_    16
#define KC    4
#define R_    96
#define RN2   128      // R + 2N
#define LTILES 13      // ceil(196/16)

typedef float v2f __attribute__((ext_vector_type(2)));
typedef float v8f __attribute__((ext_vector_type(8)));

__device__ __forceinline__ float sigmoid_(float x) {
    return 1.0f / (1.0f + __expf(-x));
}

// ---------------------------------------------------------------------------
// Kernel 1: in_proj GEMM via V_WMMA_F32_16X16X4_F32.
//   xz[b, e, t, l] = sum_d hidden[b,t,l,d] * in_proj_w[e,d]
//   e < E  -> x_pre[b, c=e*T+t, l]   ;  e >= E -> z[b, c=(e-E)*T+t, l]
// One wave per 16(l) x 16(e) tile; K loop over 192 with two independent
// WMMA accumulator chains (step 8). Out-of-range l rows are CLAMPED (not
// masked): garbage A row m only affects D row m, which is never stored.
// ---------------------------------------------------------------------------
__global__ __launch_bounds__(32)
void k_gemm_inproj(const float* __restrict__ hidden,
                   const float* __restrict__ w,       // (2E, D)
                   float* __restrict__ x_pre,
                   float* __restrict__ zbuf)
{
    const int lane = threadIdx.x;
    const int et   = blockIdx.x % 24;      // 384/16 e-tiles
    const int lt   = blockIdx.x / 24;      // 13 l-tiles
    const int bt   = blockIdx.y;           // b*T + t
    const int b    = bt / T_;
    const int t    = bt % T_;

    const int mrow  = lt * 16 + (lane & 15);          // l row for A fragment
    const int mclmp = (mrow < L_) ? mrow : (L_ - 1);  // clamp, no mask needed
    const int koff  = (lane >> 4) * 2;                // lanes 16-31: K+2,K+3
    const int ecol  = et * 16 + (lane & 15);          // e col for B fragment

    const float* Arow = hidden + ((size_t)bt * L_ + mclmp) * D_;
    const float* Brow = w + (size_t)ecol * D_;

    v8f acc0 = {0.f,0.f,0.f,0.f,0.f,0.f,0.f,0.f};
    v8f acc1 = {0.f,0.f,0.f,0.f,0.f,0.f,0.f,0.f};
    for (int k0 = 0; k0 < D_; k0 += 8) {
        v2f a0 = *(const v2f*)(Arow + k0 + koff);
        v2f b0 = *(const v2f*)(Brow + k0 + koff);
        v2f a1 = *(const v2f*)(Arow + k0 + 4 + koff);
        v2f b1 = *(const v2f*)(Brow + k0 + 4 + koff);
        acc0 = __builtin_amdgcn_wmma_f32_16x16x4_f32(
                   false, a0, false, b0, (short)0, acc0, false, false);
        acc1 = __builtin_amdgcn_wmma_f32_16x16x4_f32(
                   false, a1, false, b1, (short)0, acc1, false, false);
    }

    // D layout: VGPR r -> M = r + 8*(lane>=16), N = lane&15
    const int nn    = lane & 15;
    const int e_out = et * 16 + nn;
    const int mbase = (lane >> 4) * 8;
    #pragma unroll
    for (int r = 0; r < 8; ++r) {
        const int l = lt * 16 + mbase + r;
        if (l < L_) {
            const float v = acc0[r] + acc1[r];
            if (e_out < E_) {
                const int c = e_out * T_ + t;
                x_pre[((size_t)b * C_ + c) * L_ + l] = v;
            } else {
                const int c = (e_out - E_) * T_ + t;
                zbuf[((size_t)b * C_ + c) * L_ + l] = v;
            }
        }
    }
}

// ---------------------------------------------------------------------------
// Kernel 2: depthwise causal conv (K=4) + bias + SiLU.
// ---------------------------------------------------------------------------
__global__ __launch_bounds__(256)
void k_conv_silu(const float* __restrict__ x_pre,
                 const float* __restrict__ cw,   // (C, K)
                 const float* __restrict__ cb,   // (C,)
                 float* __restrict__ xc)
{
    const int idx = blockIdx.x * 256 + threadIdx.x;
    if (idx >= BSZ * C_ * L_) return;
    const int l  = idx % L_;
    const int bc = idx / L_;
    const int c  = bc % C_;

    const float* xp = x_pre + (size_t)bc * L_;
    float s = cb[c];
    #pragma unroll
    for (int k = 0; k < KC; ++k) {
        const int ls = l - (KC - 1) + k;
        if (ls >= 0) s += xp[ls] * cw[c * KC + k];
    }
    xc[idx] = s * sigmoid_(s);
}

// ---------------------------------------------------------------------------
// Kernel 3: x_proj GEMM. x_dbl[b, r, l] = sum_c x_proj_w[r,c] * xc[b,c,l]
// M=128 (r), N=196 (l), K=1536. Out-of-range l columns clamped (garbage
// D column n never stored). Two WMMA chains.
// ---------------------------------------------------------------------------
__global__ __launch_bounds__(32)
void k_gemm_xproj(const float* __restrict__ xc,
                  const float* __restrict__ w,       // (128, C)
                  float* __restrict__ x_dbl)
{
    const int lane = threadIdx.x;
    const int rt   = blockIdx.x % 8;     // 128/16 r-tiles
    const int lt   = blockIdx.x / 8;     // 13 l-tiles
    const int b    = blockIdx.y;

    const int rrow  = rt * 16 + (lane & 15);
    const int koff  = (lane >> 4) * 2;
    const int lcol  = lt * 16 + (lane & 15);
    const int lclmp = (lcol < L_) ? lcol : (L_ - 1);

    const float* Arow  = w + (size_t)rrow * C_;
    const float* Bcol  = xc + (size_t)b * C_ * L_ + lclmp;   // + k*L_ per row

    v8f acc0 = {0.f,0.f,0.f,0.f,0.f,0.f,0.f,0.f};
    v8f acc1 = {0.f,0.f,0.f,0.f,0.f,0.f,0.f,0.f};
    for (int k0 = 0; k0 < C_; k0 += 8) {
        const int kB = k0 + koff;
        v2f a0 = *(const v2f*)(Arow + k0 + koff);
        v2f b0; b0.x = Bcol[(size_t)kB * L_];
                b0.y = Bcol[(size_t)(kB + 1) * L_];
        v2f a1 = *(const v2f*)(Arow + k0 + 4 + koff);
        v2f b1; b1.x = Bcol[(size_t)(kB + 4) * L_];
                b1.y = Bcol[(size_t)(kB + 5) * L_];
        acc0 = __builtin_amdgcn_wmma_f32_16x16x4_f32(
                   false, a0, false, b0, (short)0, acc0, false, false);
        acc1 = __builtin_amdgcn_wmma_f32_16x16x4_f32(
                   false, a1, false, b1, (short)0, acc1, false, false);
    }

    const int nn    = lane & 15;
    const int l     = lt * 16 + nn;
    const int mbase = (lane >> 4) * 8;
    if (l < L_) {
        #pragma unroll
        for (int r = 0; r < 8; ++r) {
            const int rr = rt * 16 + mbase + r;
            x_dbl[((size_t)b * RN2 + rr) * L_ + l] = acc0[r] + acc1[r];
        }
    }
}

// ---------------------------------------------------------------------------
// Kernel 4: dt_proj GEMM + bias + softplus.
//   delta[b,c,l] = softplus( sum_r dt_proj_w[c,r] * x_dbl[b,r,l] + dt_bias[c] )
// M=1536 (c), N=196 (l), K=96.
// ---------------------------------------------------------------------------
__global__ __launch_bounds__(32)
void k_gemm_dtproj(const float* __restrict__ x_dbl,
                   const float* __restrict__ w,       // (C, R)
                   const float* __restrict__ dt_bias, // (C,)
                   float* __restrict__ delta)
{
    const int lane = threadIdx.x;
    const int ct   = blockIdx.x % 96;    // 1536/16 c-tiles
    const int lt   = blockIdx.x / 96;    // 13 l-tiles
    const int b    = blockIdx.y;

    const int crow  = ct * 16 + (lane & 15);
    const int koff  = (lane >> 4) * 2;
    const int lcol  = lt * 16 + (lane & 15);
    const int lclmp = (lcol < L_) ? lcol : (L_ - 1);

    const float* Arow = w + (size_t)crow * R_;
    const float* Bcol = x_dbl + (size_t)b * RN2 * L_ + lclmp;  // rows 0..95 = dt

    v8f acc0 = {0.f,0.f,0.f,0.f,0.f,0.f,0.f,0.f};
    v8f acc1 = {0.f,0.f,0.f,0.f,0.f,0.f,0.f,0.f};
    for (int k0 = 0; k0 < R_; k0 += 8) {
        const int kB = k0 + koff;
        v2f a0 = *(const v2f*)(Arow + k0 + koff);
        v2f b0; b0.x = Bcol[(size_t)kB * L_];
                b0.y = Bcol[(size_t)(kB + 1) * L_];
        v2f a1 = *(const v2f*)(Arow + k0 + 4 + koff);
        v2f b1; b1.x = Bcol[(size_t)(kB + 4) * L_];
                b1.y = Bcol[(size_t)(kB + 5) * L_];
        acc0 = __builtin_amdgcn_wmma_f32_16x16x4_f32(
                   false, a0, false, b0, (short)0, acc0, false, false);
        acc1 = __builtin_amdgcn_wmma_f32_16x16x4_f32(
                   false, a1, false, b1, (short)0, acc1, false, false);
    }

    const int nn    = lane & 15;
    const int l     = lt * 16 + nn;
    const int mbase = (lane >> 4) * 8;
    if (l < L_) {
        #pragma unroll
        for (int r = 0; r < 8; ++r) {
            const int c = ct * 16 + mbase + r;
            const float v = acc0[r] + acc1[r] + dt_bias[c];
            const float sp = (v > 20.f) ? v : __logf(1.f + __expf(v));
            delta[((size_t)b * C_ + c) * L_ + l] = sp;
        }
    }
}

// ---------------------------------------------------------------------------
// Kernel 5: fused selective scan + output.
// 16 lanes per (b,c) scan (lane = state index n); 2 scans per wave32.
//   h_n <- h_n * exp(delta * A_n) + delta * B_n(l) * x
//   y(l) = sum_n h_n * C_n(l);  out = (y + x*D) * silu(z)
// ---------------------------------------------------------------------------
__global__ __launch_bounds__(256)
void k_scan_out(const float* __restrict__ delta,
                const float* __restrict__ xc,
                const float* __restrict__ zbuf,
                const float* __restrict__ x_dbl,
                const float* __restrict__ A_log,    // (C, N)
                const float* __restrict__ Dp_,      // (C,)
                float* __restrict__ out)
{
    const int g  = blockIdx.x * 256 + threadIdx.x;   // 0 .. B*C*16-1
    const int n  = g & 15;
    const int bc = g >> 4;
    const int b  = bc / C_;
    const int c  = bc % C_;

    const float A_neg = -__expf(A_log[c * N_ + n]);
    const float Dp    = Dp_[c];

    const float* dp = delta + (size_t)bc * L_;
    const float* xp = xc    + (size_t)bc * L_;
    const float* zp = zbuf  + (size_t)bc * L_;
    const float* Bs = x_dbl + ((size_t)b * RN2 + R_       + n) * L_;
    const float* Cs = x_dbl + ((size_t)b * RN2 + R_ + N_  + n) * L_;
    float* op = out + (size_t)bc * L_;

    float h = 0.f;
    for (int l = 0; l < L_; ++l) {
        const float dv = dp[l];
        const float xv = xp[l];
        const float dA = __expf(dv * A_neg);
        h = h * dA + dv * Bs[l] * xv;
        float yp = h * Cs[l];
        yp += __shfl_xor(yp, 1, 16);
        yp += __shfl_xor(yp, 2, 16);
        yp += __shfl_xor(yp, 4, 16);
        yp += __shfl_xor(yp, 8, 16);
        if (n == 0) {
            const float zv = zp[l];
            op[l] = (yp + xv * Dp) * (zv * sigmoid_(zv));
        }
    }
}

// ---------------------------------------------------------------------------
extern "C" void kernel_launch(void* const* d_in, const int* in_sizes, int n_in,
                              void* d_out, int out_size, void* d_ws, size_t ws_size,
                              hipStream_t stream)
{
    const float* hidden    = (const float*)d_in[0];
    const float* in_proj_w = (const float*)d_in[1];
    const float* conv_w    = (const float*)d_in[2];
    const float* conv_b    = (const float*)d_in[3];
    const float* x_proj_w  = (const float*)d_in[4];
    const float* dt_proj_w = (const float*)d_in[5];
    const float* dt_bias   = (const float*)d_in[6];
    const float* A_log     = (const float*)d_in[7];
    const float* D_param   = (const float*)d_in[8];

    float* ws = (float*)d_ws;
    const size_t BCL = (size_t)BSZ * C_ * L_;   // 2,408,448
    float* x_pre = ws;
    float* zbuf  = ws + BCL;
    float* xcbuf = ws + 2 * BCL;
    float* delta = ws + 3 * BCL;
    float* x_dbl = ws + 4 * BCL;                // B*128*L floats

    // 1) in_proj GEMM (WMMA f32): 24 e-tiles x 13 l-tiles, per (b,t)
    k_gemm_inproj<<<dim3(24 * LTILES, BSZ * T_), 32, 0, stream>>>(
        hidden, in_proj_w, x_pre, zbuf);

    // 2) depthwise conv + bias + SiLU
    k_conv_silu<<<(BSZ * C_ * L_ + 255) / 256, 256, 0, stream>>>(
        x_pre, conv_w, conv_b, xcbuf);

    // 3) x_proj GEMM (WMMA f32): 8 r-tiles x 13 l-tiles, per b
    k_gemm_xproj<<<dim3(8 * LTILES, BSZ), 32, 0, stream>>>(
        xcbuf, x_proj_w, x_dbl);

    // 4) dt_proj GEMM + softplus (WMMA f32): 96 c-tiles x 13 l-tiles, per b
    k_gemm_dtproj<<<dim3(96 * LTILES, BSZ), 32, 0, stream>>>(
        x_dbl, dt_proj_w, dt_bias, delta);

    // 5) fused scan + gating output: B*C*N lanes
    k_scan_out<<<(BSZ * C_ * N_) / 256, 256, 0, stream>>>(
        delta, xcbuf, zbuf, x_dbl, A_log, D_param, (float*)d_out);
}